// AntisymmetricExpGenerator_54065048322368
// MI455X (gfx1250) — compile-verified
//
#include <hip/hip_runtime.h>
#include <hip/hip_bf16.h>

#define H_DIM 2048
#define UDU 512
#define Y_DIM 512
#define DELTA_F 0.01f
#define NCHUNK 4                 // K-dimension split per strip (one chunk per wave)
#define CHUNK_LEN (H_DIM / NCHUNK)

typedef __attribute__((ext_vector_type(2))) float v2f;
typedef __attribute__((ext_vector_type(8))) float v8f;

__device__ inline float waveReduce(float v) {
    #pragma unroll
    for (int off = 16; off > 0; off >>= 1) v += __shfl_xor(v, off, 32);
    return v;
}

// ---------------------------------------------------------------------------
// init: cur_h = h, acc_rec = h ; b = B_w @ cat(du,u), cur_b = acc_inp = DELTA*b
// one wave per row of B_w (2048 rows). grid=256, block=256 (8 waves).
// ---------------------------------------------------------------------------
__global__ __launch_bounds__(256) void init_kernel(
    const float* __restrict__ u, const float* __restrict__ du,
    const float* __restrict__ h, const float* __restrict__ Bw,
    float* __restrict__ cur,   // [2*H]: h-series then b-series
    float* __restrict__ acc)   // [2*H]: rec then inp
{
    const int lane = threadIdx.x & 31;
    const int wave = threadIdx.x >> 5;
    const int row  = blockIdx.x * 8 + wave;          // 0..2047
    float sum = 0.0f;
    for (int j = lane; j < UDU; j += 32) {
        float c = (j < 256) ? du[j] : u[j - 256];    // cat([du, u])
        sum += Bw[(size_t)row * UDU + j] * c;
    }
    sum = waveReduce(sum);
    if (lane == 0) {
        float hv = h[row];
        cur[row] = hv;
        acc[row] = hv;
        float q0 = DELTA_F * sum;                    // j=0 term of Delta*phi1 series
        cur[H_DIM + row] = q0;
        acc[H_DIM + row] = q0;
    }
}

// ---------------------------------------------------------------------------
// dual_matvec: per-chunk partials of  W @ [vh vb]  and  W^T @ [vh vb]
// via V_WMMA_F32_16X16X4_F32.
// grid = 256 blocks x 128 threads (4 waves). Blocks 0..127: 16-row strips
// (W v). Blocks 128..255: 16-col strips (W^T v). Wave w of a block reduces
// K-chunk [w*512, w*512+512) and writes partial result slice [chunk][2H].
// Loops use constant trip counts + pointer bumping so loop control stays
// scalar (s_cmp/s_cbranch) and EXEC remains all-ones around the WMMAs.
// ---------------------------------------------------------------------------
__global__ __launch_bounds__(128) void dual_matvec(
    const float* __restrict__ W,
    const float* __restrict__ cur,   // [2*H]: vh then vb
    float* __restrict__ yrp,         // [NCHUNK][2*H] partials of W  @ v
    float* __restrict__ ycp)         // [NCHUNK][2*H] partials of W^T @ v
{
    __shared__ __align__(16) float ldsv[2 * H_DIM];
    const int tid = threadIdx.x;
    for (int i = tid; i < (2 * H_DIM) / 4; i += blockDim.x)
        ((float4*)ldsv)[i] = ((const float4*)cur)[i];
    __syncthreads();

    const int lane  = tid & 31;
    const int chunk = tid >> 5;                  // 0..3 : K-chunk handled by wave
    const int j0    = chunk * CHUNK_LEN;
    const int n     = lane & 15;                 // WMMA N index / M index of A tile
    const int ko    = (lane >> 4) * 2;           // K offset held by this half-wave
    const int voff  = (n == 0) ? 0 : H_DIM;      // col 0 = h-series, col 1 = b-series
    const bool bact = (n < 2);

    v8f accv = {};
    const float* lp = &ldsv[voff + j0 + ko];     // staged vector values for this lane

    if (blockIdx.x < 128) {
        // ---- row mode: rows r0..r0+15, reduce over columns j in this chunk
        const int r0 = blockIdx.x * 16;
        // A[m=n, k=ko..ko+1] tile source for this lane
        const float* wp = W + (size_t)(r0 + n) * H_DIM + ko + j0;
        #pragma unroll 4
        for (int it = 0; it < CHUNK_LEN / 4; ++it) {
            v2f a = *(const v2f*)wp;             // 8B-aligned b64 load
            v2f t = *(const v2f*)lp;
            v2f b;
            b.x = bact ? t.x : 0.0f;             // B[k=ko  ][n] = v_n[j+ko]
            b.y = bact ? t.y : 0.0f;             // B[k=ko+1][n] = v_n[j+ko+1]
            accv = __builtin_amdgcn_wmma_f32_16x16x4_f32(
                false, a, false, b, (short)0, accv, false, false);
            wp += 4;
            lp += 4;
        }
        // D: lane N holds M=0..7 in v0..7 (lanes 0..15) or M=8..15 (lanes 16..31)
        if (bact) {
            float* dst = yrp + (size_t)chunk * 2 * H_DIM + voff + r0 + (lane >> 4) * 8;
            #pragma unroll
            for (int g = 0; g < 8; ++g) dst[g] = accv[g];
        }
    } else {
        // ---- col mode: outputs are columns c0..c0+15, reduce over rows r.
        // A'[m,k] = W[r+k, c0+m]  (4x16 slab of W transposed into A layout)
        const int c0 = (blockIdx.x - 128) * 16;
        const float* wp = W + ((size_t)ko + j0) * H_DIM + c0 + n;
        #pragma unroll 2
        for (int it = 0; it < CHUNK_LEN / 4; ++it) {
            v2f a;
            a.x = wp[0];                          // W[r+ko  , c0+n]
            a.y = wp[H_DIM];                      // W[r+ko+1, c0+n]
            v2f t = *(const v2f*)lp;
            v2f b;
            b.x = bact ? t.x : 0.0f;              // B'[k][n] = v_n[r+k]
            b.y = bact ? t.y : 0.0f;
            accv = __builtin_amdgcn_wmma_f32_16x16x4_f32(
                false, a, false, b, (short)0, accv, false, false);
            wp += 4 * H_DIM;
            lp += 4;
        }
        if (bact) {
            float* dst = ycp + (size_t)chunk * 2 * H_DIM + voff + c0 + (lane >> 4) * 8;
            #pragma unroll
            for (int g = 0; g < 8; ++g) dst[g] = accv[g];
        }
    }
}

// ---------------------------------------------------------------------------
// combine partials: t_k = (0.5*D/k)     * sum_c (yrp_h - ycp_h)
//                   q_k = (0.5*D/(k+1)) * sum_c (yrp_b - ycp_b)
// cur = new term; acc += new term.
// ---------------------------------------------------------------------------
__global__ __launch_bounds__(256) void combine_kernel(
    const float* __restrict__ yrp, const float* __restrict__ ycp,
    float* __restrict__ cur, float* __restrict__ acc, float sh, float sb)
{
    const int i = blockIdx.x * blockDim.x + threadIdx.x;   // 0..H-1
    float dh = 0.0f, db = 0.0f;
    #pragma unroll
    for (int c = 0; c < NCHUNK; ++c) {
        const size_t o = (size_t)c * 2 * H_DIM;
        dh += yrp[o + i]         - ycp[o + i];
        db += yrp[o + H_DIM + i] - ycp[o + H_DIM + i];
    }
    float th = sh * dh;
    cur[i] = th;
    acc[i] += th;
    float tb = sb * db;
    cur[H_DIM + i] = tb;
    acc[H_DIM + i] += tb;
}

// h_new = acc_rec + acc_inp
__global__ __launch_bounds__(256) void finalize_h(
    const float* __restrict__ acc, float* __restrict__ hnew)
{
    const int i = blockIdx.x * blockDim.x + threadIdx.x;
    hnew[i] = acc[i] + acc[H_DIM + i];
}

// y = C_w @ h_new : one wave per output row (512). grid=64, block=256 (8 waves)
__global__ __launch_bounds__(256) void gemv_y(
    const float* __restrict__ Cw, const float* __restrict__ hnew,
    float* __restrict__ y)
{
    const int lane = threadIdx.x & 31;
    const int wave = threadIdx.x >> 5;
    const int o = blockIdx.x * 8 + wave;          // 0..511
    float sum = 0.0f;
    for (int j = lane; j < H_DIM; j += 32)
        sum += Cw[(size_t)o * H_DIM + j] * hnew[j];
    sum = waveReduce(sum);
    if (lane == 0) y[o] = sum;
}

extern "C" void kernel_launch(void* const* d_in, const int* in_sizes, int n_in,
                              void* d_out, int out_size, void* d_ws, size_t ws_size,
                              hipStream_t stream) {
    const float* u  = (const float*)d_in[0];   // [256]
    const float* du = (const float*)d_in[1];   // [256]
    const float* h  = (const float*)d_in[2];   // [2048]
    const float* W  = (const float*)d_in[3];   // [2048*2048]
    const float* Bw = (const float*)d_in[4];   // [2048*512]
    const float* Cw = (const float*)d_in[5];   // [512*2048]
    float* y  = (float*)d_out;                 // [512]
    float* ws = (float*)d_ws;

    float* cur  = ws;                         // [2*H]
    float* acc  = ws + 2 * H_DIM;             // [2*H]
    float* yrp  = ws + 4 * H_DIM;             // [NCHUNK][2*H]
    float* ycp  = ws + (4 + 2 * NCHUNK) * H_DIM;        // [NCHUNK][2*H]
    float* hnew = ws + (4 + 4 * NCHUNK) * H_DIM;        // [H]

    init_kernel<<<256, 256, 0, stream>>>(u, du, h, Bw, cur, acc);

    const int K = 8;   // ||DELTA*A|| ~ 0.04 -> 8 terms is far below fp32 eps
    for (int k = 1; k <= K; ++k) {
        dual_matvec<<<256, 128, 0, stream>>>(W, cur, yrp, ycp);
        combine_kernel<<<H_DIM / 256, 256, 0, stream>>>(
            yrp, ycp, cur, acc,
            0.5f * DELTA_F / (float)k,          // skew 0.5 folded in
            0.5f * DELTA_F / (float)(k + 1));
    }

    finalize_h<<<H_DIM / 256, 256, 0, stream>>>(acc, hnew);
    gemv_y<<<64, 256, 0, stream>>>(Cw, hnew, y);
}